// SphericalBesselLayer_83665962926262
// MI455X (gfx1250) — compile-verified
//
#include <hip/hip_runtime.h>
#include <math.h>

#define NSPH 7
#define NRAD 6
#define NCOL (NSPH * NRAD)   // 42
#define BLOCK 256
#define R_CUT 5.0f

typedef __attribute__((ext_vector_type(2))) float v2f;
typedef __attribute__((ext_vector_type(8))) float v8f;

struct BesselTab {
    float zeros[NCOL];   // zeros[l*6+n] of spherical bessel j_l
    float rnorm[NCOL];   // 1/sqrt(0.5*j_{l+1}(z)^2)
};

// sph_norm[l] = sqrt((2l+1)/(4*pi))
#define SN0 0.28209479177387814
#define SN1 0.48860251190291992
#define SN2 0.63078313050504009
#define SN3 0.74635266518023080
#define SN4 0.84628437532163443
#define SN5 0.93560257962738882
#define SN6 1.01710723628205458

// B matrix for D = A * B : rows K = power of cos, cols N = l (>=7 zero).
// B[k][l] = sph_norm[l] * (coefficient of c^k in P_l(c))
__device__ __constant__ float BT_c[8][16] = {
  { (float)(SN0), 0.f, (float)(-0.5*SN2), 0.f, (float)(0.375*SN4), 0.f, (float)(-0.3125*SN6),
    0.f, 0.f,0.f,0.f,0.f,0.f,0.f,0.f,0.f },
  { 0.f, (float)(SN1), 0.f, (float)(-1.5*SN3), 0.f, (float)(1.875*SN5), 0.f,
    0.f, 0.f,0.f,0.f,0.f,0.f,0.f,0.f,0.f },
  { 0.f, 0.f, (float)(1.5*SN2), 0.f, (float)(-3.75*SN4), 0.f, (float)(6.5625*SN6),
    0.f, 0.f,0.f,0.f,0.f,0.f,0.f,0.f,0.f },
  { 0.f, 0.f, 0.f, (float)(2.5*SN3), 0.f, (float)(-8.75*SN5), 0.f,
    0.f, 0.f,0.f,0.f,0.f,0.f,0.f,0.f,0.f },
  { 0.f, 0.f, 0.f, 0.f, (float)(4.375*SN4), 0.f, (float)(-19.6875*SN6),
    0.f, 0.f,0.f,0.f,0.f,0.f,0.f,0.f,0.f },
  { 0.f, 0.f, 0.f, 0.f, 0.f, (float)(7.875*SN5), 0.f,
    0.f, 0.f,0.f,0.f,0.f,0.f,0.f,0.f,0.f },
  { 0.f, 0.f, 0.f, 0.f, 0.f, 0.f, (float)(14.4375*SN6),
    0.f, 0.f,0.f,0.f,0.f,0.f,0.f,0.f,0.f },
  { 0.f, 0.f, 0.f, 0.f, 0.f, 0.f, 0.f,
    0.f, 0.f,0.f,0.f,0.f,0.f,0.f,0.f,0.f }
};

__global__ __launch_bounds__(BLOCK)
void SphericalBesselLayer_83665962926262_kernel(
    const float* __restrict__ dist,   // [nedge]
    const float* __restrict__ ang,    // [ntrip]
    const float* __restrict__ amask,  // [ntrip]
    const int*   __restrict__ kj,     // [ntrip] expand_to_kj
    float*       __restrict__ out,    // [ntrip*42]
    int ntrip, int nedge, BesselTab tb)
{
    __shared__ float rbf_lds[BLOCK * NCOL];  // 42 KB: per-triplet radial row (mask*env folded)
    __shared__ float sbf_lds[BLOCK * 8];     //  8 KB: per-triplet 7 angular values (padded)

    const int tid = threadIdx.x;
    const long long base = (long long)blockIdx.x * BLOCK;
    long long tt = base + tid;
    // Clamp instead of early-exit so EXEC stays all-ones for WMMA in tail block.
    const int t = (tt < (long long)ntrip) ? (int)tt : (ntrip - 1);

    // Streamed once -> non-temporal loads (preserve L2 for the dist gather table).
    int e = __builtin_nontemporal_load(kj + t);
    e = (e < 0) ? 0 : ((e >= nedge) ? (nedge - 1) : e);
    const float d  = dist[e];      // 4 MB table, 3x reuse -> regular (L2-resident) load
    const float mk = __builtin_nontemporal_load(amask + t);
    const float a  = __builtin_nontemporal_load(ang + t);

    // ---- envelope: 1 - 28 s^6 + 48 s^7 - 21 s^8, zero for s >= 1 ----
    const float s  = d * (1.0f / R_CUT);
    const float s2 = s * s;
    const float s6 = s2 * s2 * s2;
    float env = 1.0f + s6 * (-28.0f + s * (48.0f - 21.0f * s));
    env = (s < 1.0f) ? env : 0.0f;
    const float envm = env * mk;   // fold angle_mask into the radial row

    // ---- radial basis: norm[l,n] * j_l(s * z[l,n]) * env * mask -> LDS ----
    #pragma unroll
    for (int l = 0; l < NSPH; ++l) {
        #pragma unroll
        for (int n = 0; n < NRAD; ++n) {
            const int idx = l * NRAD + n;
            const float u = s * tb.zeros[idx];
            float sn, cs;
            __sincosf(u, &sn, &cs);
            const float inv = 1.0f / u;
            const float j0 = sn * inv;
            float j;
            if (l == 0) {
                j = j0;
            } else {
                float jm1 = j0;
                float jc  = (sn * inv - cs) * inv;    // j1
                #pragma unroll
                for (int m = 1; m < l; ++m) {
                    const float nj = (2.0f * (float)m + 1.0f) * inv * jc - jm1;
                    jm1 = jc; jc = nj;
                }
                j = jc;
            }
            rbf_lds[tid * NCOL + idx] = tb.rnorm[idx] * j * envm;
        }
    }

    // ---- angular basis via V_WMMA_F32_16X16X4_F32 (exact f32 matmul) ----
    // sbf[l] = sph_norm[l]*P_l(c) = sum_k c^k * BT[k][l], degrees 0..7 via 2 chained WMMAs.
    {
        const float c = __cosf(a);
        const unsigned lane = (unsigned)tid & 31u;
        const bool hi = lane >= 16u;              // A layout: lanes 0-15 hold K=0,1; 16-31 hold K=2,3
        const int  lsel = (int)(lane & 15u);
        const int  warpBase = tid & ~31;

        v2f B1, B2;                                // B 4x16: lanes 0-15 rows K,K+1; 16-31 rows K+2,K+3
        B1.x = hi ? BT_c[2][lsel] : BT_c[0][lsel];
        B1.y = hi ? BT_c[3][lsel] : BT_c[1][lsel];
        B2.x = hi ? BT_c[6][lsel] : BT_c[4][lsel];
        B2.y = hi ? BT_c[7][lsel] : BT_c[5][lsel];

        #pragma unroll
        for (int half = 0; half < 2; ++half) {     // 16 triplets per WMMA pair, 32 per wave
            const float cm  = __shfl(c, (half << 4) + lsel, 32);
            const float cm2 = cm * cm;
            const float cm4 = cm2 * cm2;
            v2f A1, A2;
            A1.x = hi ? cm2       : 1.0f;          // K = 2 : 0
            A1.y = hi ? cm2 * cm  : cm;            // K = 3 : 1
            A2.x = hi ? cm4 * cm2 : cm4;           // K = 6 : 4
            A2.y = hi ? 0.0f      : cm4 * cm;      // K = 7 (B row 7 == 0) : 5
            v8f acc = {};
            acc = __builtin_amdgcn_wmma_f32_16x16x4_f32(
                false, A1, false, B1, (short)0, acc, false, false);
            acc = __builtin_amdgcn_wmma_f32_16x16x4_f32(
                false, A2, false, B2, (short)0, acc, false, false);
            // D layout: VGPR r -> rows r (lanes 0-15) / r+8 (lanes 16-31), col N = lane&15
            if (lsel < NSPH) {
                const int trBase = warpBase + (half << 4) + (hi ? 8 : 0);
                #pragma unroll
                for (int r = 0; r < 8; ++r)
                    sbf_lds[(trBase + r) * 8 + lsel] = acc[r];
            }
        }
    }

    __syncthreads();

    // ---- fused multiply + coalesced streaming (NT) b64 stores ----
    // Pair base index is even and totalOut (= ntrip*42) is even, so each pair is
    // entirely in-bounds or entirely out-of-bounds; 8-byte alignment guaranteed.
    const long long totalOut = (long long)ntrip * NCOL;
    const long long outBase  = base * NCOL;
    #pragma unroll
    for (int jj = 0; jj < NCOL / 2; ++jj) {        // 21 iterations, 2 elems/thread
        const int idx0 = (jj * BLOCK + tid) * 2;   // 0 .. 256*42-2, contiguous per wave
        const int idx1 = idx0 + 1;
        const int r0 = idx0 / NCOL;
        const int c0 = idx0 - r0 * NCOL;
        const int r1 = idx1 / NCOL;
        const int c1 = idx1 - r1 * NCOL;
        v2f v;
        v.x = rbf_lds[r0 * NCOL + c0] * sbf_lds[r0 * 8 + c0 / NRAD];
        v.y = rbf_lds[r1 * NCOL + c1] * sbf_lds[r1 * 8 + c1 / NRAD];
        const long long o = outBase + idx0;
        if (o < totalOut)
            __builtin_nontemporal_store(v, (v2f*)(out + o));   // keep 504MB stream out of L2
    }
}

// ---------------- host side ----------------

static double jn_d(int l, double x) {
    const double j0 = sin(x) / x;
    if (l == 0) return j0;
    double jm1 = j0;
    double j = sin(x) / (x * x) - cos(x) / x;
    for (int m = 1; m < l; ++m) {
        const double nj = (2.0 * m + 1.0) / x * j - jm1;
        jm1 = j; j = nj;
    }
    return j;
}

extern "C" void kernel_launch(void* const* d_in, const int* in_sizes, int n_in,
                              void* d_out, int out_size, void* d_ws, size_t ws_size,
                              hipStream_t stream) {
    const float* dist  = (const float*)d_in[0];  // pair_distances
    const float* ang   = (const float*)d_in[1];  // angles
    const float* amask = (const float*)d_in[2];  // angle_mask
    // d_in[3] = reduce_to_ji: unused by the reference
    const int*   kj    = (const int*)d_in[4];    // expand_to_kj
    const int nedge = in_sizes[0];
    const int ntrip = in_sizes[1];
    if (ntrip <= 0 || nedge <= 0) return;

    // Bessel zeros + norms: same double-precision bisection as the reference.
    BesselTab tb;
    const double PI = 3.14159265358979323846;
    double zer[NSPH][NRAD];
    {
        double points[NSPH + NRAD - 1];
        double rac[NSPH + NRAD - 1];
        for (int k = 0; k < NRAD; ++k) zer[0][k] = (k + 1) * PI;
        for (int j = 0; j < NSPH + NRAD - 1; ++j) { points[j] = (j + 1) * PI; rac[j] = 0.0; }
        for (int i = 1; i < NSPH; ++i) {
            const int cnt = NSPH + NRAD - 1 - i;
            for (int j = 0; j < cnt; ++j) {
                double a = points[j], b = points[j + 1];
                double fa = jn_d(i, a);
                for (int it = 0; it < 100; ++it) {
                    const double m = 0.5 * (a + b);
                    const double fm = jn_d(i, m);
                    if (fa * fm <= 0.0) b = m; else { a = m; fa = fm; }
                }
                rac[j] = 0.5 * (a + b);
            }
            for (int j = 0; j < NSPH + NRAD - 1; ++j) points[j] = rac[j];
            for (int k = 0; k < NRAD; ++k) zer[i][k] = rac[k];
        }
    }
    for (int l = 0; l < NSPH; ++l)
        for (int n = 0; n < NRAD; ++n) {
            tb.zeros[l * NRAD + n] = (float)zer[l][n];
            const double jv = jn_d(l + 1, zer[l][n]);
            tb.rnorm[l * NRAD + n] = (float)(1.0 / sqrt(0.5 * jv * jv));
        }

    const int nblk = (ntrip + BLOCK - 1) / BLOCK;
    SphericalBesselLayer_83665962926262_kernel<<<nblk, BLOCK, 0, stream>>>(
        dist, ang, amask, kj, (float*)d_out, ntrip, nedge, tb);
}